// Scalar3DHMM_14121852469705
// MI455X (gfx1250) — compile-verified
//
#include <hip/hip_runtime.h>
#include <stdint.h>

// Problem constants (fixed by the reference)
#define Vc     10000
#define Nc     16384      // 32*32*16, power of two
#define Bc     32
#define Tc     64
#define Lc     16
#define ROWS   4          // emission rows staged in LDS per workgroup
#define K3T    256

typedef __attribute__((ext_vector_type(4))) unsigned int v4u;
typedef __attribute__((ext_vector_type(8))) int v8i;
typedef __attribute__((ext_vector_type(4))) int v4i;

// ---------------------------------------------------------------------------
// CDNA5 async global->LDS copy (16B per lane) + wait  (fallback path)
// ---------------------------------------------------------------------------
__device__ __forceinline__ void async_copy16(uint32_t lds_byte_off, uint64_t gaddr) {
  asm volatile("global_load_async_to_lds_b128 %0, %1, off"
               :: "v"(lds_byte_off), "v"(gaddr) : "memory");
}
__device__ __forceinline__ void wait_async0() {
  asm volatile("s_wait_asynccnt 0" ::: "memory");
}

// ---------------------------------------------------------------------------
// CDNA5 Tensor Data Mover: one instruction stages 4 contiguous emission rows
// (4 x 10000 x 4B = 160000 B) from global into LDS.  D# per ISA ch.8.
// This toolchain exposes the 6-arg builtin:
//   (uint32x4 g0, int32x8 g1, int32x4 g2, int32x4 g3, int32x8 extra, i32 cpol)
// ---------------------------------------------------------------------------
#if __has_builtin(__builtin_amdgcn_tensor_load_to_lds)
#define HAVE_TDM 1
__device__ __forceinline__ void tdm_load_rows(uint32_t lds_addr, uint64_t gaddr) {
  v4u g0;
  g0.x = 1u;                                         // count=1 (valid), user mode
  g0.y = lds_addr;                                   // lds_addr [63:32]
  g0.z = (uint32_t)(gaddr & 0xFFFFFFFFu);            // global_addr low
  g0.w = (uint32_t)((gaddr >> 32) & 0x01FFFFFFu)     // global_addr[56:32]
         | 0x80000000u;                              // type=2 ("image")
  v8i g1;
  g1[0] = 0x00020000;                // workgroup_mask=0, data_size=2 (4B)
  g1[1] = (int)(10000u << 16);       // tensor_dim0 low16  (=10000)
  g1[2] = (int)(4u << 16);           // tensor_dim0 hi16=0 | tensor_dim1 low16 (=4)
  g1[3] = (int)(10000u << 16);       // tensor_dim1 hi16=0 | tile_dim0 (=10000)
  g1[4] = 4;                         // tile_dim1=4, tile_dim2=0 (unused)
  g1[5] = 10000;                     // tensor_dim0_stride low32 (elements)
  g1[6] = 0;                         // stride hi + tensor_dim1_stride low (unused)
  g1[7] = 0;
  v4i z4; z4[0] = 0; z4[1] = 0; z4[2] = 0; z4[3] = 0;   // groups 2/3 unused
  v8i z8; z8[0] = 0; z8[1] = 0; z8[2] = 0; z8[3] = 0;
          z8[4] = 0; z8[5] = 0; z8[6] = 0; z8[7] = 0;
  __builtin_amdgcn_tensor_load_to_lds(g0, g1, z4, z4, z8, 0);
}
#else
#define HAVE_TDM 0
#endif

// ---------------------------------------------------------------------------
// K1: logT[j][n] = log_softmax(trans[n, :7])[j], stored transposed (7 x N)
// ---------------------------------------------------------------------------
__global__ void k_logT(const float* __restrict__ trans, float* __restrict__ logTt) {
  int n = blockIdx.x * blockDim.x + threadIdx.x;
  if (n >= Nc) return;
  float t[7];
#pragma unroll
  for (int j = 0; j < 7; ++j) t[j] = trans[n * 7 + j];
  float m = t[0];
#pragma unroll
  for (int j = 1; j < 7; ++j) m = fmaxf(m, t[j]);
  float s = 0.f;
#pragma unroll
  for (int j = 0; j < 7; ++j) s += __expf(t[j] - m);
  float lse = m + __logf(s);
#pragma unroll
  for (int j = 0; j < 7; ++j) logTt[j * Nc + n] = t[j] - lse;
}

// ---------------------------------------------------------------------------
// K2: priorLSE[0] = logsumexp(prior[0..N))   (single 1024-thread block)
// ---------------------------------------------------------------------------
__global__ void k_priorlse(const float* __restrict__ prior, float* __restrict__ out1) {
  __shared__ float red[1024];
  int tid = threadIdx.x;
  float m = -INFINITY;
  for (int i = tid; i < Nc; i += 1024) m = fmaxf(m, prior[i]);
  red[tid] = m; __syncthreads();
  for (int s = 512; s > 0; s >>= 1) {
    if (tid < s) red[tid] = fmaxf(red[tid], red[tid + s]);
    __syncthreads();
  }
  float gm = red[0]; __syncthreads();
  float a = 0.f;
  for (int i = tid; i < Nc; i += 1024) a += __expf(prior[i] - gm);
  red[tid] = a; __syncthreads();
  for (int s = 512; s > 0; s >>= 1) {
    if (tid < s) red[tid] += red[tid + s];
    __syncthreads();
  }
  if (tid == 0) out1[0] = gm + __logf(red[0]);
}

// ---------------------------------------------------------------------------
// K3: emission scores.  WG handles states n0..n0+3 (n0 = 4*blockIdx).
//  - TDM (or async-to-LDS) copies 4 contiguous emis rows into LDS
//  - in-LDS logsumexp per row
//  - em[(t*B+b)*N + n] = sum_l emisLDS[n][tok(t,b,l)] - L*rowLSE[n]
// ---------------------------------------------------------------------------
__global__ void k_emis(const float* __restrict__ emis,
                       const int* __restrict__ stories,
                       float* __restrict__ em) {
  extern __shared__ float rowbuf[];          // ROWS * Vc floats (160000 B)
  __shared__ float red[K3T];
  const int n0  = blockIdx.x * ROWS;
  const int tid = threadIdx.x;

  const uint64_t gbase = (uint64_t)(uintptr_t)emis + (uint64_t)n0 * Vc * 4u;
  const uint32_t lds0  = (uint32_t)(uintptr_t)(void*)rowbuf;  // low 32b = LDS byte addr

#if HAVE_TDM
  if (tid < 32u) {                           // wave 0 issues the TDM op (wave-level)
    tdm_load_rows(lds0, gbase);
    __builtin_amdgcn_s_wait_tensorcnt(0);
  }
#else
  const int nchunks = ROWS * Vc * 4 / 16;    // 10000 x 16B chunks
  for (int c = tid; c < nchunks; c += K3T)
    async_copy16(lds0 + (uint32_t)c * 16u, gbase + (uint64_t)c * 16u);
  wait_async0();
#endif
  __syncthreads();

  // --- per-row logsumexp over V ---
  float lse[ROWS];
#pragma unroll
  for (int r = 0; r < ROWS; ++r) {
    float m = -INFINITY;
    for (int i = tid; i < Vc; i += K3T) m = fmaxf(m, rowbuf[r * Vc + i]);
    red[tid] = m; __syncthreads();
    for (int s = K3T / 2; s > 0; s >>= 1) {
      if (tid < s) red[tid] = fmaxf(red[tid], red[tid + s]);
      __syncthreads();
    }
    float gm = red[0]; __syncthreads();
    float a = 0.f;
    for (int i = tid; i < Vc; i += K3T) a += __expf(rowbuf[r * Vc + i] - gm);
    red[tid] = a; __syncthreads();
    for (int s = K3T / 2; s > 0; s >>= 1) {
      if (tid < s) red[tid] += red[tid + s];
      __syncthreads();
    }
    lse[r] = gm + __logf(red[0]);
    __syncthreads();
  }

  // --- emission accumulation for all (t,b) pairs ---
  const int TB = Tc * Bc;                    // 2048
  for (int tb = tid; tb < TB; tb += K3T) {
    const int t = tb >> 5;                   // tb / B
    const int b = tb & 31;                   // tb % B
    const int* tok = stories + ((size_t)b * Tc + t) * Lc;
    float acc[ROWS] = {0.f, 0.f, 0.f, 0.f};
#pragma unroll 4
    for (int l = 0; l < Lc; ++l) {
      const int vv = tok[l];
#pragma unroll
      for (int r = 0; r < ROWS; ++r) acc[r] += rowbuf[r * Vc + vv];
    }
    float4 o;
    o.x = acc[0] - (float)Lc * lse[0];
    o.y = acc[1] - (float)Lc * lse[1];
    o.z = acc[2] - (float)Lc * lse[2];
    o.w = acc[3] - (float)Lc * lse[3];
    *(float4*)&em[(size_t)tb * Nc + n0] = o;
  }
}

// ---------------------------------------------------------------------------
// K4: scores0[b*N+n] = em[(0*B+b)*N+n] + (prior[n] - priorLSE)   (4-wide)
// ---------------------------------------------------------------------------
__global__ void k_init(const float* __restrict__ em, const float* __restrict__ prior,
                       const float* __restrict__ priorLSE, float* __restrict__ scores) {
  int g = blockIdx.x * blockDim.x + threadIdx.x;       // one float4 per thread
  if (g >= Bc * Nc / 4) return;
  const int n0 = (g & ((Nc / 4) - 1)) << 2;
  const float pl = priorLSE[0];
  const float4 e = *(const float4*)&em[(size_t)g * 4];
  const float4 p = *(const float4*)&prior[n0];
  float4 o;
  o.x = e.x + p.x - pl;
  o.y = e.y + p.y - pl;
  o.z = e.z + p.z - pl;
  o.w = e.w + p.w - pl;
  *(float4*)&scores[(size_t)g * 4] = o;
}

// ---------------------------------------------------------------------------
// K5: one forward step, 4 states per thread (all-float4 traffic).
// dst[b,n] = em_t[b,n] + LSE_j( logTt[j][n] + src[b, (n+delta_j) & (N-1)] )
// deltas: {0, -1, +1, -32, +32, -1024, -2048}
// ---------------------------------------------------------------------------
__global__ void k_step(const float* __restrict__ src, const float* __restrict__ emt,
                       const float* __restrict__ logTt, float* __restrict__ dst) {
  const int g = blockIdx.x * blockDim.x + threadIdx.x;   // 0 .. B*N/4-1
  if (g >= Bc * Nc / 4) return;
  const int b  = g >> 12;                  // (g*4) / N
  const int n0 = (g & ((Nc / 4) - 1)) << 2;
  const float* sb = src + (size_t)b * Nc;

  // aligned neighbor blocks (groups of 4 stay contiguous under mod-N: all
  // deltas here are multiples of 4, N is a multiple of 4)
  const float4 s0   = *(const float4*)&sb[n0];
  const float4 sm32 = *(const float4*)&sb[(n0 - 32) & (Nc - 1)];
  const float4 sp32 = *(const float4*)&sb[(n0 + 32) & (Nc - 1)];
  const float4 sm1k = *(const float4*)&sb[(n0 - 1024) & (Nc - 1)];
  const float4 sm2k = *(const float4*)&sb[(n0 - 2048) & (Nc - 1)];
  const float  sm1  = sb[(n0 - 1) & (Nc - 1)];   // for delta=-1, lane 0
  const float  sp4  = sb[(n0 + 4) & (Nc - 1)];   // for delta=+1, lane 3

  float4 lt[7];
#pragma unroll
  for (int j = 0; j < 7; ++j) lt[j] = *(const float4*)&logTt[j * Nc + n0];
  const float4 e = *(const float4*)&emt[(size_t)g * 4];

  // per-lane neighbor values
  float p[7][4];
  p[0][0] = s0.x;  p[0][1] = s0.y;  p[0][2] = s0.z;  p[0][3] = s0.w;   // delta 0
  p[1][0] = sm1;   p[1][1] = s0.x;  p[1][2] = s0.y;  p[1][3] = s0.z;   // delta -1
  p[2][0] = s0.y;  p[2][1] = s0.z;  p[2][2] = s0.w;  p[2][3] = sp4;    // delta +1
  p[3][0] = sm32.x; p[3][1] = sm32.y; p[3][2] = sm32.z; p[3][3] = sm32.w;
  p[4][0] = sp32.x; p[4][1] = sp32.y; p[4][2] = sp32.z; p[4][3] = sp32.w;
  p[5][0] = sm1k.x; p[5][1] = sm1k.y; p[5][2] = sm1k.z; p[5][3] = sm1k.w;
  p[6][0] = sm2k.x; p[6][1] = sm2k.y; p[6][2] = sm2k.z; p[6][3] = sm2k.w;

  const float* ltf = (const float*)lt;     // lt[j][i] at ltf[j*4+i]
  const float* ef  = (const float*)&e;
  float o[4];
#pragma unroll
  for (int i = 0; i < 4; ++i) {
    float vt[7];
#pragma unroll
    for (int j = 0; j < 7; ++j) vt[j] = ltf[j * 4 + i] + p[j][i];
    float m = vt[0];
#pragma unroll
    for (int j = 1; j < 7; ++j) m = fmaxf(m, vt[j]);
    float s = 0.f;
#pragma unroll
    for (int j = 0; j < 7; ++j) s += __expf(vt[j] - m);
    o[i] = ef[i] + m + __logf(s);
  }
  float4 ov; ov.x = o[0]; ov.y = o[1]; ov.z = o[2]; ov.w = o[3];
  *(float4*)&dst[(size_t)g * 4] = ov;
}

// ---------------------------------------------------------------------------
// K6: out[b] = logsumexp over n of scores[b, :]   (float4 loads)
// ---------------------------------------------------------------------------
__global__ void k_final(const float* __restrict__ scores, float* __restrict__ out) {
  __shared__ float red[256];
  const int b = blockIdx.x, tid = threadIdx.x;
  const float4* s4 = (const float4*)(scores + (size_t)b * Nc);
  float m = -INFINITY;
  for (int i = tid; i < Nc / 4; i += 256) {
    float4 v = s4[i];
    m = fmaxf(m, fmaxf(fmaxf(v.x, v.y), fmaxf(v.z, v.w)));
  }
  red[tid] = m; __syncthreads();
  for (int k = 128; k > 0; k >>= 1) {
    if (tid < k) red[tid] = fmaxf(red[tid], red[tid + k]);
    __syncthreads();
  }
  float gm = red[0]; __syncthreads();
  float a = 0.f;
  for (int i = tid; i < Nc / 4; i += 256) {
    float4 v = s4[i];
    a += __expf(v.x - gm) + __expf(v.y - gm) + __expf(v.z - gm) + __expf(v.w - gm);
  }
  red[tid] = a; __syncthreads();
  for (int k = 128; k > 0; k >>= 1) {
    if (tid < k) red[tid] += red[tid + k];
    __syncthreads();
  }
  if (tid == 0) out[b] = gm + __logf(red[0]);
}

// ---------------------------------------------------------------------------
// Host launcher
// ---------------------------------------------------------------------------
extern "C" void kernel_launch(void* const* d_in, const int* in_sizes, int n_in,
                              void* d_out, int out_size, void* d_ws, size_t ws_size,
                              hipStream_t stream) {
  (void)in_sizes; (void)n_in; (void)out_size; (void)ws_size;
  const int*   stories = (const int*)d_in[0];
  const float* trans   = (const float*)d_in[3];
  const float* emis    = (const float*)d_in[4];
  const float* prior   = (const float*)d_in[5];
  float* out = (float*)d_out;

  // workspace layout (floats)
  float* ws       = (float*)d_ws;
  float* em       = ws;                                   // T*B*N = 33.5M floats
  float* logTt    = em + (size_t)Tc * Bc * Nc;            // 7*N
  float* priorLSE = logTt + (size_t)7 * Nc;               // 1 (+pad)
  float* buf0     = priorLSE + 16;                        // B*N
  float* buf1     = buf0 + (size_t)Bc * Nc;               // B*N

  const size_t shbytes = (size_t)ROWS * Vc * sizeof(float);  // 160000 B
  (void)hipFuncSetAttribute((const void*)k_emis,
                            hipFuncAttributeMaxDynamicSharedMemorySize,
                            (int)shbytes);

  k_logT    <<<Nc / 256, 256, 0, stream>>>(trans, logTt);
  k_priorlse<<<1, 1024, 0, stream>>>(prior, priorLSE);
  k_emis    <<<Nc / ROWS, K3T, shbytes, stream>>>(emis, stories, em);
  k_init    <<<(Bc * Nc / 4) / 256, 256, 0, stream>>>(em, prior, priorLSE, buf0);

  float* cur = buf0;
  float* nxt = buf1;
  for (int t = 1; t < Tc; ++t) {
    k_step<<<(Bc * Nc / 4) / 256, 256, 0, stream>>>(cur, em + (size_t)t * Bc * Nc, logTt, nxt);
    float* tmp = cur; cur = nxt; nxt = tmp;
  }
  k_final<<<Bc, 256, 0, stream>>>(cur, out);
}